// PreMSA_20194936225854
// MI455X (gfx1250) — compile-verified
//
#include <hip/hip_runtime.h>
#include <hip/hip_bf16.h>

#define L_SEQ   768
#define N_MSA   256
#define SEQ_DIM 22
#define MSA_DIM 23
#define M_DIM   256
#define Z_DIM   128
#define NBITS   14
#define RELCLIP 32

typedef __attribute__((ext_vector_type(2))) float v2f;
typedef __attribute__((ext_vector_type(4))) float v4f;
typedef __attribute__((ext_vector_type(8))) float v8f;

// ---------------------------------------------------------------------------
// Kernel 1: tiny precompute of all row-broadcast terms (L2-resident inputs).
//   s_plus_p[l,c] = (seq@Ws)[l,c] + bs[c] + (bits(l)@Wpos2)[l,c] + bpos2[c] + bmsa[c]
//   sqp[l,c]      = (seq@Wq)[l,c] + bq[c] + bpos[c]
//   sk [l,c]      = (seq@Wk)[l,c] + bk[c]
// ---------------------------------------------------------------------------
__global__ void premsa_precompute(const float* __restrict__ seq,
                                  const float* __restrict__ Ws,  const float* __restrict__ bs,
                                  const float* __restrict__ bmsa,
                                  const float* __restrict__ Wq,  const float* __restrict__ bq,
                                  const float* __restrict__ Wk,  const float* __restrict__ bk,
                                  const float* __restrict__ Wpos2, const float* __restrict__ bpos2,
                                  const float* __restrict__ bpos,
                                  float* __restrict__ s_plus_p,   // [L, M_DIM]
                                  float* __restrict__ sqp,        // [L, Z_DIM]
                                  float* __restrict__ sk)         // [L, Z_DIM]
{
    const int t  = blockIdx.x * blockDim.x + threadIdx.x;
    const int T0 = L_SEQ * M_DIM;
    const int T1 = T0 + L_SEQ * Z_DIM;
    const int T2 = T1 + L_SEQ * Z_DIM;
    if (t < T0) {
        const int l = t / M_DIM, c = t % M_DIM;
        float acc = bs[c] + bpos2[c] + bmsa[c];
#pragma unroll
        for (int k = 0; k < SEQ_DIM; ++k) acc += seq[l * SEQ_DIM + k] * Ws[k * M_DIM + c];
#pragma unroll
        for (int j = 0; j < NBITS; ++j)
            if ((l >> j) & 1) acc += Wpos2[j * M_DIM + c];
        s_plus_p[t] = acc;
    } else if (t < T1) {
        const int t2 = t - T0;
        const int l = t2 / Z_DIM, c = t2 % Z_DIM;
        float acc = bq[c] + bpos[c];
#pragma unroll
        for (int k = 0; k < SEQ_DIM; ++k) acc += seq[l * SEQ_DIM + k] * Wq[k * Z_DIM + c];
        sqp[t2] = acc;
    } else if (t < T2) {
        const int t2 = t - T1;
        const int l = t2 / Z_DIM, c = t2 % Z_DIM;
        float acc = bk[c];
#pragma unroll
        for (int k = 0; k < SEQ_DIM; ++k) acc += seq[l * SEQ_DIM + k] * Wk[k * Z_DIM + c];
        sk[t2] = acc;
    }
}

// ---------------------------------------------------------------------------
// Kernel 2: m = msa @ Wmsa (+ fused s_plus_p bias), via V_WMMA_F32_16X16X4_F32.
// One block = one 16-row strip of the flattened [N*L, 256] output.
// 8 waves; each wave owns two 16x16 column tiles (32 cols). K=23 padded to 24
// -> 6 WMMA steps of K=4 per tile (12 wmma per wave).
// A frag (16x4 f32): lane&15 = M, VGPR v holds K = v + 2*(lane>>4).
// B frag (4x16 f32): lane&15 = N, VGPR v holds K = v + 2*(lane>>4).
// C/D (16x16 f32):   VGPR v -> row v + 8*(lane>>4), col = lane&15.
// Padding handled branch-free: clamp address in-bounds, cndmask value to 0.
// Output stored non-temporally (write-once stream, 201 MB > L2).
// ---------------------------------------------------------------------------
__global__ void premsa_m_wmma(const float* __restrict__ msa,
                              const float* __restrict__ Wmsa,
                              const float* __restrict__ s_plus_p,
                              float* __restrict__ m_out)
{
    const int row0 = blockIdx.x * 16;        // flattened row = n*L + l
    const int l0   = row0 % L_SEQ;           // uniform, once per block (SALU)
    const int wave = threadIdx.x >> 5;       // 0..7
    const int lane = threadIdx.x & 31;
    const int mrow = lane & 15;
    const int hi   = lane >> 4;
    const int col0 = wave * 32;

    v8f c0 = {};
    v8f c1 = {};
    const float* __restrict__ arow = msa + (size_t)(row0 + mrow) * MSA_DIM;

#pragma unroll
    for (int kk = 0; kk < 24; kk += 4) {
        v2f a, b0, b1;
#pragma unroll
        for (int v = 0; v < 2; ++v) {
            const int  k  = kk + v + 2 * hi;
            const bool ok = (k < MSA_DIM);
            const int  ks = ok ? k : 0;      // clamped, always in-bounds
            float av  = arow[ks];
            float bv0 = Wmsa[ks * M_DIM + col0 + mrow];
            float bv1 = Wmsa[ks * M_DIM + col0 + 16 + mrow];
            a[v]  = ok ? av  : 0.0f;         // v_cndmask, no exec branching
            b0[v] = ok ? bv0 : 0.0f;
            b1[v] = ok ? bv1 : 0.0f;
        }
        c0 = __builtin_amdgcn_wmma_f32_16x16x4_f32(false, a, false, b0, (short)0, c0, false, false);
        c1 = __builtin_amdgcn_wmma_f32_16x16x4_f32(false, a, false, b1, (short)0, c1, false, false);
    }

#pragma unroll
    for (int v = 0; v < 8; ++v) {
        const int r  = v + 8 * hi;
        const int gr = row0 + r;
        const int l  = l0 + r;               // no per-row division
        const float bias0 = s_plus_p[l * M_DIM + col0 + mrow];
        const float bias1 = s_plus_p[l * M_DIM + col0 + 16 + mrow];
        __builtin_nontemporal_store(c0[v] + bias0, &m_out[(size_t)gr * M_DIM + col0 + mrow]);
        __builtin_nontemporal_store(c1[v] + bias1, &m_out[(size_t)gr * M_DIM + col0 + 16 + mrow]);
    }
}

// ---------------------------------------------------------------------------
// Kernel 3: z[i,j,:] = sqp[j,:] + sk[i,:] + Wpos[clip(j-i)+32,:]
// Pure streaming NT store, float4 per thread, fully coalesced.
// ---------------------------------------------------------------------------
__global__ void premsa_z(const float* __restrict__ sqp,
                         const float* __restrict__ sk,
                         const float* __restrict__ Wpos,
                         float* __restrict__ z_out)
{
    const int i  = blockIdx.y;
    const int e  = blockIdx.x * blockDim.x + threadIdx.x;  // float4-group index in row i
    const int j  = e >> 5;                                 // Z_DIM/4 == 32 groups per j
    const int c4 = (e & 31) * 4;

    int rel = j - i;
    rel = rel < -RELCLIP ? -RELCLIP : (rel > RELCLIP ? RELCLIP : rel);
    rel += RELCLIP;

    const v4f q = *(const v4f*)(sqp + j * Z_DIM + c4);
    const v4f k = *(const v4f*)(sk + i * Z_DIM + c4);
    const v4f p = *(const v4f*)(Wpos + rel * Z_DIM + c4);
    __builtin_nontemporal_store(q + k + p,
        (v4f*)(z_out + ((size_t)i * L_SEQ + j) * Z_DIM + c4));
}

// ---------------------------------------------------------------------------
extern "C" void kernel_launch(void* const* d_in, const int* in_sizes, int n_in,
                              void* d_out, int out_size, void* d_ws, size_t ws_size,
                              hipStream_t stream) {
    const float* seq   = (const float*)d_in[0];
    const float* msa   = (const float*)d_in[1];
    const float* Wmsa  = (const float*)d_in[2];
    const float* bmsa  = (const float*)d_in[3];
    const float* Ws    = (const float*)d_in[4];
    const float* bs    = (const float*)d_in[5];
    const float* Wq    = (const float*)d_in[6];
    const float* bq    = (const float*)d_in[7];
    const float* Wk    = (const float*)d_in[8];
    const float* bk    = (const float*)d_in[9];
    const float* Wpos  = (const float*)d_in[10];
    const float* bpos  = (const float*)d_in[11];
    const float* Wpos2 = (const float*)d_in[12];
    const float* bpos2 = (const float*)d_in[13];

    float* m_out = (float*)d_out;                                  // [N, L, M_DIM]
    float* z_out = m_out + (size_t)N_MSA * L_SEQ * M_DIM;          // [L, L, Z_DIM]

    float* s_plus_p = (float*)d_ws;                                // [L, M_DIM]
    float* sqp      = s_plus_p + L_SEQ * M_DIM;                    // [L, Z_DIM]
    float* skb      = sqp + L_SEQ * Z_DIM;                         // [L, Z_DIM]

    const int totPre = L_SEQ * M_DIM + 2 * L_SEQ * Z_DIM;
    premsa_precompute<<<(totPre + 255) / 256, 256, 0, stream>>>(
        seq, Ws, bs, bmsa, Wq, bq, Wk, bk, Wpos2, bpos2, bpos, s_plus_p, sqp, skb);

    premsa_m_wmma<<<(N_MSA * L_SEQ) / 16, 256, 0, stream>>>(msa, Wmsa, s_plus_p, m_out);

    dim3 zgrid((L_SEQ * (Z_DIM / 4)) / 256, L_SEQ);
    premsa_z<<<zgrid, 256, 0, stream>>>(sqp, skb, Wpos, z_out);
}